// Skeletonize_IBP_48954037239852
// MI455X (gfx1250) — compile-verified
//
#include <hip/hip_runtime.h>
#include <math.h>

// Skeletonize (iterative boundary peeling), CDNA5 / gfx1250.
// x is stored as a guarded byte image: 1028 rows x 1040 byte pitch.
//   bits 0..3 : binary pixel for batches 0..3
//   bit  4    : endpoint flag (batch-summed ring count <= 1), refreshed per iteration
// Padded coords: (r,c) in [1,1026]^2 <-> x-coords [0,1025]^2. Guard ring = 0.

#define XPITCH 1040
#define XROWS  1028
#define XBYTES (XROWS * XPITCH)        // 1,069,120 bytes (~1.05 MB, L2 resident)
#define EPSF   1e-8f
#define BETAF  0.33f

#define TI 16                           // output sites per tile (rows)
#define TJ 64                           // output sites per tile (cols)
#define TROWS (2 * TI + 1)              // 33 window rows
#define TPITCH 144                      // LDS tile pitch: 129 needed + 16B-align slack (9 x b128)
#define TILE_BYTES (TROWS * TPITCH)     // 4752

#if defined(__has_builtin)
# if __has_builtin(__builtin_amdgcn_global_load_async_to_lds_b128)
#  define HAVE_ASYNC_B128 1
# endif
# if __has_builtin(__builtin_amdgcn_global_load_async_to_lds_b32)
#  define HAVE_ASYNC_B32 1
# endif
#endif

typedef int v4i __attribute__((vector_size(16)));
typedef __attribute__((address_space(1))) v4i gas_v4i;
typedef __attribute__((address_space(3))) v4i las_v4i;
typedef __attribute__((address_space(1))) int gas_i32;
typedef __attribute__((address_space(3))) int las_i32;

// ---------------- discretize: img+noise -> packed binary bytes -----------------
__global__ void k_discretize(const float* __restrict__ img,
                             const float* __restrict__ noise,
                             unsigned char* __restrict__ xb) {
  int idx = blockIdx.x * blockDim.x + threadIdx.x;
  if (idx >= XBYTES) return;
  int r = idx / XPITCH, c = idx - r * XPITCH;
  unsigned byte = 0u;
  if (r >= 1 && r <= 1026 && c >= 1 && c <= 1026) {
    int xr = r - 1, xc = c - 1;                 // x coords in [0,1025]
    bool inner = (xr >= 1 && xr <= 1024 && xc >= 1 && xc <= 1024);
#pragma unroll
    for (int b = 0; b < 4; ++b) {
      float im = inner ? img[(b * 1024 + (xr - 1)) * 1024 + (xc - 1)] : 0.0f;
      float nz = noise[(b * 1026 + xr) * 1026 + xc];
      float z = logf((im + EPSF) / (1.0f - im + EPSF))
              + BETAF * (logf(nz) - logf(1.0f - nz));
      float soft = 1.0f / (1.0f + expf(-z));
      if (soft > 0.5f) byte |= (1u << b);
    }
  }
  xb[idx] = (unsigned char)byte;
}

// ---------------- endpoint pass: set bit4 = (batch-summed ring count <= 1) ------
__global__ void k_endpoint(unsigned char* __restrict__ xb) {
  int q = blockIdx.x * blockDim.x + threadIdx.x;
  if (q >= 1026 * 1026) return;
  int r = 1 + q / 1026, c = 1 + (q % 1026);
  const int base = r * XPITCH + c;
  int n = 0;
  n += __popc(xb[base - XPITCH - 1] & 0xF);
  n += __popc(xb[base - XPITCH    ] & 0xF);
  n += __popc(xb[base - XPITCH + 1] & 0xF);
  n += __popc(xb[base          - 1] & 0xF);
  n += __popc(xb[base          + 1] & 0xF);
  n += __popc(xb[base + XPITCH - 1] & 0xF);
  n += __popc(xb[base + XPITCH    ] & 0xF);
  n += __popc(xb[base + XPITCH + 1] & 0xF);
  unsigned char me = xb[base] & 0xF;            // races on neighbor reads are benign: bits0-3 never change here
  if (n <= 1) me |= 0x10;
  xb[base] = me;
}

__device__ __forceinline__ float face_t(float q) {
  return fmaxf(0.25f * q - 0.5f, 0.0f) * 2.0f;
}

// ---------------- subfield peel pass (xo,yo) ------------------------------------
// Tile of TI x TJ sites; window = (2TI+1) x (2TJ+1) bytes staged to LDS via
// async global->LDS DMA (CDNA5 path, b128 chunks), then ds_load_u8 window reads.
__global__ __launch_bounds__(256) void k_subfield(unsigned char* __restrict__ xb,
                                                  int xo, int yo) {
  __shared__ __align__(16) unsigned char tile[TILE_BYTES];
  const int tid = threadIdx.x;
  const int i0 = min((int)blockIdx.y * TI, 513 - TI);   // overlapping clamp: idempotent (clears only)
  const int j0 = min((int)blockIdx.x * TJ, 513 - TJ);
  const int R0 = xo + 2 * i0;                           // padded row of window top
  const int C0 = yo + 2 * j0;                           // padded col of window left
  const int C0a = C0 & ~15;                             // 16B-aligned load base (max col touched = 1039)
  const int shift = C0 - C0a;                           // 0..15
  const unsigned char* gbase = xb + R0 * XPITCH + C0a;

#if defined(HAVE_ASYNC_B128)
  for (int t = tid; t < TROWS * (TPITCH / 16); t += 256) {   // 297 x b128
    int row = t / (TPITCH / 16), ch = t - row * (TPITCH / 16);
    const unsigned char* g = gbase + row * XPITCH + ch * 16;
    __builtin_amdgcn_global_load_async_to_lds_b128((gas_v4i*)g, (las_v4i*)&tile[t * 16], 0, 0);
  }
#elif defined(HAVE_ASYNC_B32)
  for (int t = tid; t < TROWS * (TPITCH / 4); t += 256) {    // 1188 x b32
    int row = t / (TPITCH / 4), dw = t - row * (TPITCH / 4);
    const unsigned char* g = gbase + row * XPITCH + dw * 4;
    __builtin_amdgcn_global_load_async_to_lds_b32((gas_i32*)g, (las_i32*)&tile[t * 4], 0, 0);
  }
#else
  for (int t = tid; t < TROWS * (TPITCH / 4); t += 256) {
    int row = t / (TPITCH / 4), dw = t - row * (TPITCH / 4);
    const unsigned char* g = gbase + row * XPITCH + dw * 4;
    *(unsigned int*)&tile[t * 4] = *(const unsigned int*)g;
  }
#endif
#if defined(HAVE_ASYNC_B128) || defined(HAVE_ASYNC_B32)
# if __has_builtin(__builtin_amdgcn_s_wait_asynccnt)
  __builtin_amdgcn_s_wait_asynccnt(0);
# else
  asm volatile("s_wait_asynccnt 0" ::: "memory");
# endif
#endif
  __syncthreads();

#pragma unroll
  for (int k = 0; k < 4; ++k) {
    int site = tid + k * 256;                  // 0..1023 -> (si,sj) in 16x64
    int si = site >> 6, sj = site & 63;
    int lr = 2 * si, lc = shift + 2 * sj;      // lc+2 <= 15+126+2 = 143 < TPITCH

    unsigned char w11 = 0;
    float s[3][3];
#pragma unroll
    for (int du = 0; du < 3; ++du)
#pragma unroll
      for (int dv = 0; dv < 3; ++dv) {
        unsigned char wb = tile[(lr + du) * TPITCH + (lc + dv)];
        if (du == 1 && dv == 1) w11 = wb;
        s[du][dv] = 4.0f - 2.0f * (float)__popc(wb & 0xF);
      }
    float sc = s[1][1];

    // dE = (edges with masked center, s_c -> 4) - (edges with s_c); only 4 terms differ.
    float dE = 0.0f;
    {
      float o;
      o = s[0][1]; dE += (2.0f + 0.5f * o) - fmaxf(0.5f * (sc + o), 0.0f);
      o = s[2][1]; dE += (2.0f + 0.5f * o) - fmaxf(0.5f * (sc + o), 0.0f);
      o = s[1][0]; dE += (2.0f + 0.5f * o) - fmaxf(0.5f * (sc + o), 0.0f);
      o = s[1][2]; dE += (2.0f + 0.5f * o) - fmaxf(0.5f * (sc + o), 0.0f);
    }
    // dF: all 4 face blocks contain the center.
    float q00 = s[0][0] + s[0][1] + s[1][0] + s[1][1];
    float q01 = s[0][1] + s[0][2] + s[1][1] + s[1][2];
    float q10 = s[1][0] + s[1][1] + s[2][0] + s[2][1];
    float q11 = s[1][1] + s[1][2] + s[2][1] + s[2][2];
    float dF = (face_t(q00 - sc + 4.0f) - face_t(q00))
             + (face_t(q01 - sc + 4.0f) - face_t(q01))
             + (face_t(q10 - sc + 4.0f) - face_t(q10))
             + (face_t(q11 - sc + 4.0f) - face_t(q11));

    // Batches with center bit 0 never change; batches with bit 1 share Delta = 1 - dE + dF.
    float d1 = 1.0f - dE + dF;
    bool simple1 = fabsf(d1) < 0.5f;           // change < 0.5  (exact: multiples of 0.5)
    bool ep = (w11 >> 4) & 1;
    bool clear = simple1 && !ep;

    unsigned char nb = clear ? (unsigned char)(w11 & 0x10)
                             : (unsigned char)(w11 & 0x1F);
    xb[(R0 + lr + 1) * XPITCH + (C0 + 2 * sj + 1)] = nb;   // disjoint centers: race-free
  }
}

// ---------------- expand packed bits -> float output ----------------------------
__global__ void k_expand(const unsigned char* __restrict__ xb, float* __restrict__ out) {
  int q = blockIdx.x * blockDim.x + threadIdx.x;
  if (q >= 4 * 1024 * 1024) return;
  int b = q >> 20;
  int rem = q & 0xFFFFF;
  int i = rem >> 10, j = rem & 1023;
  unsigned char v = xb[(i + 2) * XPITCH + (j + 2)];   // interior crop [1,1024]^2 -> padded +2
  out[q] = (float)((v >> b) & 1);
}

extern "C" void kernel_launch(void* const* d_in, const int* in_sizes, int n_in,
                              void* d_out, int out_size, void* d_ws, size_t ws_size,
                              hipStream_t stream) {
  const float* img   = (const float*)d_in[0];   // (4,1,1024,1024) f32
  const float* noise = (const float*)d_in[1];   // (4,1026,1026)   f32
  float* out = (float*)d_out;                   // (4,1,1024,1024) f32
  unsigned char* xb = (unsigned char*)d_ws;     // needs XBYTES (~1.05 MB)
  (void)in_sizes; (void)n_in; (void)out_size; (void)ws_size;

  dim3 blk(256);
  k_discretize<<<dim3((XBYTES + 255) / 256), blk, 0, stream>>>(img, noise, xb);

  static const int offs[4][2] = {{0, 0}, {1, 0}, {0, 1}, {1, 1}};  // (xo=row, yo=col)
  for (int it = 0; it < 5; ++it) {
    k_endpoint<<<dim3((1026 * 1026 + 255) / 256), blk, 0, stream>>>(xb);
    for (int p = 0; p < 4; ++p) {
      k_subfield<<<dim3(9, 33), blk, 0, stream>>>(xb, offs[p][0], offs[p][1]);
    }
  }
  k_expand<<<dim3((4 * 1024 * 1024 + 255) / 256), blk, 0, stream>>>(xb, out);
}